// HintProcessor_63110249447438
// MI455X (gfx1250) — compile-verified
//
#include <hip/hip_runtime.h>
#include <hip/hip_bf16.h>

typedef __attribute__((ext_vector_type(2))) float v2f;
typedef __attribute__((ext_vector_type(8))) float v8f;

#define HALF_DIM 256   // label-embedding width and box-embedding width
#define HID      512   // MLP hidden width
#define BATCH    4096
#define EMB      512   // output row width
#define T_TILES  4     // token tiles (of 16) staged in LDS per pass
#define T_STAGE  (T_TILES * 16)          // 64 tokens per pass
#define HSTRIDE  516   // LDS row stride: even (8B-aligned b64) and ==4 mod 64
                       // -> 32 lanes' ds_load_b64 hit all 64 banks conflict-free

__device__ __forceinline__ int lower_bound_i(const int* __restrict__ seg, int n, int val) {
    int lo = 0, hi = n;
    while (lo < hi) {
        int mid = (lo + hi) >> 1;
        if (seg[mid] < val) lo = mid + 1; else hi = mid;
    }
    return lo;
}

__global__ __launch_bounds__(256)
void hint_fused_kernel(const int* __restrict__ labels,
                       const int* __restrict__ seg_labels,
                       const float* __restrict__ boxes,
                       const int* __restrict__ seg_boxes,
                       const float* __restrict__ emb_table,
                       const float* __restrict__ W1,
                       const float* __restrict__ b1,
                       const float* __restrict__ W2,
                       const float* __restrict__ b2,
                       float* __restrict__ out,
                       int n_tok)
{
    const int img = blockIdx.x;      // one block per image
    const int tid = threadIdx.x;     // 256 threads = 8 waves (wave32)

    __shared__ int   bounds[4];
    __shared__ float lds_box[T_STAGE * 4];
    __shared__ float lds_h[T_STAGE * HSTRIDE];

    if (tid < 4) {
        const int* seg = (tid < 2) ? seg_labels : seg_boxes;
        bounds[tid] = lower_bound_i(seg, n_tok, img + (tid & 1));
    }
    __syncthreads();
    const int lo1 = bounds[0], hi1 = bounds[1];
    const int lo2 = bounds[2], hi2 = bounds[3];

    // ---------------- label-embedding mean: output cols [0, 256) ----------------
    {
        float acc = 0.f;
        #pragma unroll 4
        for (int i = lo1; i < hi1; ++i) {
            int lbl = labels[i];
            acc += emb_table[lbl * HALF_DIM + tid];
        }
        int cnt = hi1 - lo1;
        float rcp = (cnt > 0) ? (1.0f / (float)cnt) : 0.f;
        out[img * EMB + tid] = acc * rcp;
    }

    // ---------------- box MLP mean via fp32 WMMA: output cols [256, 512) --------
    const int lane = tid & 31;
    const int wave = tid >> 5;      // 0..7 -> covers cols [32*wave, 32*wave+32)
    const int mrow = lane & 15;     // A-matrix M index (token row within tile)
    const int kp   = lane >> 4;     // K-pair select (0: K0/K1, 1: K2/K3)

    v8f acc0 = {0.f,0.f,0.f,0.f,0.f,0.f,0.f,0.f};
    v8f acc1 = {0.f,0.f,0.f,0.f,0.f,0.f,0.f,0.f};

    for (int t0 = lo2; t0 < hi2; t0 += T_STAGE) {
        // stage up to 64 tokens' boxes (exactly one value per thread)
        {
            int t = t0 + (tid >> 2);
            lds_box[tid] = (t < hi2) ? boxes[t * 4 + (tid & 3)] : 0.f;
        }
        __syncthreads();

        // h = relu(box @ W1 + b1) into LDS; rows past segment end -> 0
        for (int i = tid; i < T_STAGE * HID; i += 256) {
            int tr = i >> 9;           // token row 0..63
            int j  = i & (HID - 1);    // hidden index 0..511
            float hv = 0.f;
            if (t0 + tr < hi2) {
                hv = b1[j]
                   + lds_box[tr*4+0] * W1[0*HID + j]
                   + lds_box[tr*4+1] * W1[1*HID + j]
                   + lds_box[tr*4+2] * W1[2*HID + j]
                   + lds_box[tr*4+3] * W1[3*HID + j];
                hv = fmaxf(hv, 0.f);
            }
            lds_h[tr * HSTRIDE + j] = hv;
        }
        __syncthreads();

        // GEMM: one B fetch feeds T_TILES WMMAs per column tile.
        // All token tiles accumulate into the same C (row-summed at the end).
        const int n0 = wave * 32 + (lane & 15);
        for (int k0 = 0; k0 < HID; k0 += 4) {
            const int ka = k0 + 2 * kp;            // this lane's K pair
            v2f bA, bB;
            bA.x = W2[(ka    ) * HALF_DIM + n0];
            bA.y = W2[(ka + 1) * HALF_DIM + n0];
            bB.x = W2[(ka    ) * HALF_DIM + n0 + 16];
            bB.y = W2[(ka + 1) * HALF_DIM + n0 + 16];
            #pragma unroll
            for (int tt = 0; tt < T_TILES; ++tt) {
                v2f a = *(const v2f*)(&lds_h[(tt * 16 + mrow) * HSTRIDE + ka]);
                acc0 = __builtin_amdgcn_wmma_f32_16x16x4_f32(
                           false, a, false, bA, (short)0, acc0, false, false);
                acc1 = __builtin_amdgcn_wmma_f32_16x16x4_f32(
                           false, a, false, bB, (short)0, acc1, false, false);
            }
        }
        __syncthreads();
    }

    // Reduce over M (all tokens of the segment), apply mean + bias.
    const int cnt2 = hi2 - lo2;
    const float rcp2 = (cnt2 > 0) ? (1.0f / (float)cnt2) : 0.f;
    #pragma unroll
    for (int t = 0; t < 2; ++t) {
        v8f a = t ? acc1 : acc0;
        // lane holds M = {v + 8*kp : v in 0..7} for its N = lane&15
        float partial = a[0]+a[1]+a[2]+a[3]+a[4]+a[5]+a[6]+a[7];
        partial += __shfl_xor(partial, 16, 32);   // merge the two M-halves
        if (lane < 16) {
            int col = wave * 32 + t * 16 + lane;
            float bias = (cnt2 > 0) ? b2[col] : 0.f;
            out[img * EMB + HALF_DIM + col] = partial * rcp2 + bias;
        }
    }
}

extern "C" void kernel_launch(void* const* d_in, const int* in_sizes, int n_in,
                              void* d_out, int out_size, void* d_ws, size_t ws_size,
                              hipStream_t stream) {
    const int*   labels     = (const int*)  d_in[0];
    const int*   seg_labels = (const int*)  d_in[1];
    const float* boxes      = (const float*)d_in[2];
    const int*   seg_boxes  = (const int*)  d_in[3];
    const float* emb_table  = (const float*)d_in[4];
    const float* W1         = (const float*)d_in[5];
    const float* b1         = (const float*)d_in[6];
    const float* W2         = (const float*)d_in[7];
    const float* b2         = (const float*)d_in[8];
    float* out = (float*)d_out;
    const int n_tok = in_sizes[0];

    hipLaunchKernelGGL(hint_fused_kernel, dim3(BATCH), dim3(256), 0, stream,
                       labels, seg_labels, boxes, seg_boxes,
                       emb_table, W1, b1, W2, b2, out, n_tok);
}